// SvmLoss_50749333570240
// MI455X (gfx1250) — compile-verified
//
#include <hip/hip_runtime.h>
#include <math.h>

#define BS 384
#define DIM 256
#define GAMMA_C 0.1f
#define REG_C 0.1f
#define CCLIP 1.0f
#define ETA_C 0.001f
#define NUM_ITER 100
#define STOP_TOL 0.01f

#define GEMM_BLOCKS 144          // 6 x 24 tile-blocks, 4 waves (tiles) each
#define ELEM_BLOCKS ((BS * BS) / 256)  // 576, exact

typedef float v2f __attribute__((ext_vector_type(2)));
typedef float v8f __attribute__((ext_vector_type(8)));

// ---------------------------------------------------------------------------
// Row L2-normalization: 768 blocks (384 xis rows + 384 xjs rows), 256 threads
// (= DIM) per block, LDS tree reduction of sum of squares.
// ---------------------------------------------------------------------------
__global__ void norm_rows_kernel(const float* __restrict__ xis,
                                 const float* __restrict__ xjs,
                                 float* __restrict__ zis,
                                 float* __restrict__ zjs) {
  __shared__ float red[256];
  int row = blockIdx.x;
  const float* src = (row < BS) ? xis : xjs;
  float* dst = (row < BS) ? zis : zjs;
  int r = (row < BS) ? row : row - BS;
  float v = src[r * DIM + threadIdx.x];
  red[threadIdx.x] = v * v;
  __syncthreads();
  for (int s = 128; s > 0; s >>= 1) {
    if (threadIdx.x < s) red[threadIdx.x] += red[threadIdx.x + s];
    __syncthreads();
  }
  dst[r * DIM + threadIdx.x] = v / sqrtf(red[0]);
}

// ---------------------------------------------------------------------------
// Scatter alpha_init (BS x BS-1) into "column" layout A[r,b] with zero diag;
// clip(relu(x),0,1). Also resets the stop flag and last-block counter.
// A[r,b] (r != b) corresponds to alpha[b, j] with j = r if r<b else r-1.
// ---------------------------------------------------------------------------
__global__ void init_alpha_kernel(const float* __restrict__ alpha_init,
                                  float* __restrict__ A,
                                  float* __restrict__ Aprev,
                                  int* __restrict__ stop,
                                  unsigned int* __restrict__ counter) {
  int i = blockIdx.x * blockDim.x + threadIdx.x;
  if (i == 0) {
    *stop = 0;
    *counter = 0u;
  }
  if (i >= BS * BS) return;
  int r = i / BS, b = i % BS;
  float v = 0.0f;
  if (r != b) {
    int j = (r < b) ? r : r - 1;
    float x = alpha_init[b * (BS - 1) + j];
    v = fminf(fmaxf(x, 0.0f), CCLIP);
  }
  A[i] = v;
  Aprev[i] = v;
}

// ---------------------------------------------------------------------------
// Setup GEMM (NT): G = X * Y^T, X,Y are BS x DIM row-major, f32 WMMA.
// blockIdx.z selects the epilogue:
//   z==0:  X=Y=zis  ->  M = k + 1 + reg*I,  KKd0 = k with zero diagonal
//   z==1:  X=zis, Y=zjs ->  Ks = k
// with k = exp(-gamma * max(0, 2 - 2 g)).
// One wave per 16x16 tile.  V_WMMA_F32_16X16X4_F32 fragment layout:
//   A frag lane: {X[m, k+2h], X[m, k+2h+1]}   (m=lane&15, h=lane>>4)
//   B frag lane: {Y[n, k+2h], Y[n, k+2h+1]}   (B[k,n] = Y[n,k], n=lane&15)
//   C/D:         acc[j] = C[j + 8h, lane&15]
// ---------------------------------------------------------------------------
__global__ void setup_gemm_kernel(const float* __restrict__ zis,
                                  const float* __restrict__ zjs,
                                  float* __restrict__ Mout,
                                  float* __restrict__ KKd0,
                                  float* __restrict__ Ks) {
  int mode = blockIdx.z;
  const float* X = zis;
  const float* Y = (mode == 0) ? zis : zjs;
  int lane = threadIdx.x & 31;
  int wave = threadIdx.x >> 5;
  int r0 = (blockIdx.x * 4 + wave) * 16;
  int c0 = blockIdx.y * 16;
  int m = lane & 15;
  int h = lane >> 4;
  const float* ap = X + (size_t)(r0 + m) * DIM + 2 * h;
  const float* bp = Y + (size_t)(c0 + m) * DIM + 2 * h;
  v8f acc = {};
#pragma unroll 8
  for (int k = 0; k < DIM; k += 4) {
    v2f a = *(const v2f*)(ap + k);
    v2f b = *(const v2f*)(bp + k);
    acc = __builtin_amdgcn_wmma_f32_16x16x4_f32(false, a, false, b,
                                                (short)0, acc, false, false);
  }
#pragma unroll
  for (int j = 0; j < 8; ++j) {
    int row = r0 + j + h * 8;
    int col = c0 + m;
    float g = acc[j];
    float sq = fmaxf(2.0f - 2.0f * g, 0.0f);
    float kv = expf(-GAMMA_C * sq);
    if (mode == 0) {
      Mout[row * BS + col] = kv + 1.0f + ((row == col) ? REG_C : 0.0f);
      KKd0[row * BS + col] = (row == col) ? 0.0f : kv;
    } else {
      Ks[row * BS + col] = kv;
    }
  }
}

// ---------------------------------------------------------------------------
// Iteration phase 1 (fused): the Nesterov z is computed ON THE FLY from
// (A, Aprev):  z[r,b] = a + beta*(a - a_prev)   (diag stays 0).
//   Blocks 0..143 : WMMA GEMM  P = M * Z  (4 waves/block, 16x16 tile/wave).
//   Blocks 144..146: per-column scalars S_b = sum_r z[r,b],
//                    T_b = sum_r KKd0[b,r]*z[r,b]  (128 threads -> 384 cols).
// ---------------------------------------------------------------------------
__global__ void iter_phase1_kernel(const int* __restrict__ stop,
                                   const float* __restrict__ Mm,
                                   const float* __restrict__ A,
                                   const float* __restrict__ Aprev,
                                   const float* __restrict__ KKd0,
                                   float beta,
                                   float* __restrict__ P,
                                   float* __restrict__ S,
                                   float* __restrict__ T) {
  if (*stop) return;
  if (blockIdx.x < GEMM_BLOCKS) {
    int bx = blockIdx.x % 6;   // tile-row group (4 tiles per block)
    int by = blockIdx.x / 6;   // tile-col
    int lane = threadIdx.x & 31;
    int wave = threadIdx.x >> 5;
    int r0 = (bx * 4 + wave) * 16;
    int c0 = by * 16;
    int m = lane & 15;
    int h = lane >> 4;
    const float* ap = Mm + (size_t)(r0 + m) * BS + 2 * h;
    int boff = (2 * h) * BS + c0 + m;
    v8f acc = {};
#pragma unroll 4
    for (int k = 0; k < BS; k += 4) {
      v2f a = *(const v2f*)(ap + k);
      int i0 = boff + k * BS;
      float a0 = A[i0];
      float p0 = Aprev[i0];
      float a1 = A[i0 + BS];
      float p1 = Aprev[i0 + BS];
      v2f b;
      b.x = a0 + beta * (a0 - p0);
      b.y = a1 + beta * (a1 - p1);
      acc = __builtin_amdgcn_wmma_f32_16x16x4_f32(false, a, false, b,
                                                  (short)0, acc, false, false);
    }
#pragma unroll
    for (int j = 0; j < 8; ++j) {
      P[(r0 + j + h * 8) * BS + c0 + m] = acc[j];
    }
  } else {
    int b = (blockIdx.x - GEMM_BLOCKS) * 128 + threadIdx.x;  // 0..383
    float s = 0.0f, t = 0.0f;
    for (int r = 0; r < BS; ++r) {
      float a = A[r * BS + b];
      float p = Aprev[r * BS + b];
      float z = a + beta * (a - p);
      s += z;
      t += KKd0[b * BS + r] * z;
    }
    S[b] = s;
    T[b] = t;
  }
}

// ---------------------------------------------------------------------------
// Iteration phase 2 (fused): PGD update + commit + convergence test.
//   grad[r,b] = P[r,b] - KKd0[r,b]*S_b - T_b - 2
//   anew      = clip(z - eta*grad, 0, 1)          (z recomputed on the fly)
//   Aprev <- A ; A <- anew                        (per-element, no hazards)
// Each block writes partial sums of (anew-a)^2 and a^2; the LAST block
// (fence + atomicAdd counter) re-reduces all partials in a fixed order
// (bitwise deterministic), sets stop if rel < tol, and resets the counter.
// ---------------------------------------------------------------------------
__global__ void iter_phase2_kernel(int* __restrict__ stop,
                                   const float* __restrict__ P,
                                   const float* __restrict__ KKd0,
                                   const float* __restrict__ S,
                                   const float* __restrict__ T,
                                   float* __restrict__ A,
                                   float* __restrict__ Aprev,
                                   float beta,
                                   float* __restrict__ partials,
                                   unsigned int* __restrict__ counter) {
  if (*stop) return;
  __shared__ float sd[256];
  __shared__ float sa[256];
  __shared__ int isLast;
  int i = blockIdx.x * blockDim.x + threadIdx.x;  // grid covers BS*BS exactly
  int r = i / BS, b = i % BS;
  float aold = A[i];
  float pold = Aprev[i];
  float anew = 0.0f;
  if (r != b) {
    float z = aold + beta * (aold - pold);
    float grad = P[i] - KKd0[i] * S[b] - T[b] - 2.0f;
    anew = fminf(fmaxf(z - ETA_C * grad, 0.0f), CCLIP);
  }
  Aprev[i] = aold;   // commit: a_prev <- a
  A[i] = anew;       // commit: a      <- a_new
  float d = anew - aold;
  sd[threadIdx.x] = d * d;
  sa[threadIdx.x] = aold * aold;
  __syncthreads();
  for (int s2 = 128; s2 > 0; s2 >>= 1) {
    if (threadIdx.x < s2) {
      sd[threadIdx.x] += sd[threadIdx.x + s2];
      sa[threadIdx.x] += sa[threadIdx.x + s2];
    }
    __syncthreads();
  }
  if (threadIdx.x == 0) {
    partials[2 * blockIdx.x] = sd[0];
    partials[2 * blockIdx.x + 1] = sa[0];
    __threadfence();
    unsigned int prev = atomicAdd(counter, 1u);
    isLast = (prev == (unsigned int)(gridDim.x - 1)) ? 1 : 0;
  }
  __syncthreads();
  if (isLast) {
    __threadfence();
    float dd = 0.0f, aa = 0.0f;
    for (int j = threadIdx.x; j < ELEM_BLOCKS; j += 256) {
      dd += partials[2 * j];
      aa += partials[2 * j + 1];
    }
    sd[threadIdx.x] = dd;
    sa[threadIdx.x] = aa;
    __syncthreads();
    for (int s2 = 128; s2 > 0; s2 >>= 1) {
      if (threadIdx.x < s2) {
        sd[threadIdx.x] += sd[threadIdx.x + s2];
        sa[threadIdx.x] += sa[threadIdx.x + s2];
      }
      __syncthreads();
    }
    if (threadIdx.x == 0) {
      float rel = sqrtf(sd[0]) / (sqrtf(sa[0]) + 1e-8f);
      if (rel < STOP_TOL) *stop = 1;
      *counter = 0u;  // re-arm for next iteration
    }
  }
}

// ---------------------------------------------------------------------------
// Final loss: out = exp( (neg - pos) / BS ) with
//   neg = sum_{r,b} A[r,b]*Ks[r,b]             (diag of A is 0)
//   pos = sum_{r,b} A[r,b]*Ks[b,b]
// ---------------------------------------------------------------------------
__global__ void loss_kernel(const float* __restrict__ A,
                            const float* __restrict__ Ks,
                            float* __restrict__ out) {
  __shared__ float sn[256];
  __shared__ float sp[256];
  float neg = 0.0f, pos = 0.0f;
  for (int i = threadIdx.x; i < BS * BS; i += 256) {
    int b = i % BS;
    float a = fminf(fmaxf(A[i], 0.0f), CCLIP);
    neg += a * Ks[i];
    pos += a * Ks[b * BS + b];
  }
  sn[threadIdx.x] = neg;
  sp[threadIdx.x] = pos;
  __syncthreads();
  for (int s = 128; s > 0; s >>= 1) {
    if (threadIdx.x < s) {
      sn[threadIdx.x] += sn[threadIdx.x + s];
      sp[threadIdx.x] += sp[threadIdx.x + s];
    }
    __syncthreads();
  }
  if (threadIdx.x == 0) out[0] = expf((sn[0] - sp[0]) / (float)BS);
}

// ---------------------------------------------------------------------------
extern "C" void kernel_launch(void* const* d_in, const int* in_sizes, int n_in,
                              void* d_out, int out_size, void* d_ws,
                              size_t ws_size, hipStream_t stream) {
  const float* xis = (const float*)d_in[0];
  const float* xjs = (const float*)d_in[1];
  const float* alpha_init = (const float*)d_in[2];
  float* out = (float*)d_out;

  // Carve ~5 MB of f32 scratch out of d_ws (256-byte aligned slices).
  float* ws = (float*)d_ws;
  size_t off = 0;
  auto carve = [&](size_t n) {
    float* p = ws + off;
    off += (n + 63) & ~(size_t)63;
    return p;
  };
  float* zis   = carve((size_t)BS * DIM);
  float* zjs   = carve((size_t)BS * DIM);
  float* Mm    = carve((size_t)BS * BS);
  float* KKd0  = carve((size_t)BS * BS);
  float* Ks    = carve((size_t)BS * BS);
  float* A     = carve((size_t)BS * BS);
  float* Aprev = carve((size_t)BS * BS);
  float* P     = carve((size_t)BS * BS);
  float* S     = carve(BS);
  float* T     = carve(BS);
  float* partials = carve(2 * ELEM_BLOCKS);
  int* stop = (int*)carve(64);
  unsigned int* counter = (unsigned int*)carve(64);
  (void)ws_size; (void)in_sizes; (void)n_in; (void)out_size;

  dim3 setupGrid(BS / 16 / 4, BS / 16, 2);  // (6, 24, 2): z = mode
  dim3 gemmBlock(128);

  // Setup: normalize rows, scatter alpha / reset flags, build M/KKd0/Ks.
  norm_rows_kernel<<<2 * BS, 256, 0, stream>>>(xis, xjs, zis, zjs);
  init_alpha_kernel<<<ELEM_BLOCKS, 256, 0, stream>>>(alpha_init, A, Aprev,
                                                     stop, counter);
  setup_gemm_kernel<<<setupGrid, gemmBlock, 0, stream>>>(zis, zjs, Mm, KKd0, Ks);

  // Fixed 100 PGD iterations, 2 launches each; every kernel gated on the
  // device stop flag so the captured graph does identical work while
  // honoring the early-exit semantics of the reference while_loop.
  for (int t = 0; t < NUM_ITER; ++t) {
    float beta = (float)t / ((float)t + 3.0f);
    iter_phase1_kernel<<<GEMM_BLOCKS + 3, 128, 0, stream>>>(
        stop, Mm, A, Aprev, KKd0, beta, P, S, T);
    iter_phase2_kernel<<<ELEM_BLOCKS, 256, 0, stream>>>(
        stop, P, KKd0, S, T, A, Aprev, beta, partials, counter);
  }

  loss_kernel<<<1, 256, 0, stream>>>(A, Ks, out);
}